// symLoss_32384053412216
// MI455X (gfx1250) — compile-verified
//
#include <hip/hip_runtime.h>

// symLoss for MI455X (gfx1250, wave32).
// Structure: grid (B, S), 256 threads/block. Each thread transforms points
// (plane reflect + quat rotate), gathers mask/closest-point, accumulates L2
// norms. Block reduction done with V_WMMA_F32_16X16X4_F32 (A = ones trick:
// D = 1*B + C column-sums 64 values per WMMA; 4 chained WMMAs reduce 256).
// Final deterministic tree-reduce kernel produces the two scalar outputs.

#define GRIDN     32
#define GRID3     32768
#define GRID_MIN_F (-0.484375f)   // -0.5 + 0.5/32

typedef __attribute__((ext_vector_type(2))) float v2f;
typedef __attribute__((ext_vector_type(8))) float v8f;

__device__ __forceinline__ float gather_norm(const float* __restrict__ cps_b,
                                             const float* __restrict__ v_b,
                                             float qx, float qy, float qz) {
    // voxel index: round(clip((p - GRID_MIN) * GRID / (2*GBOUND), 0, 31)),
    // jnp.round = round-half-even -> rintf (v_rndne_f32)
    float fx = rintf(fminf(fmaxf((qx - GRID_MIN_F) * 32.0f, 0.0f), 31.0f));
    float fy = rintf(fminf(fmaxf((qy - GRID_MIN_F) * 32.0f, 0.0f), 31.0f));
    float fz = rintf(fminf(fmaxf((qz - GRID_MIN_F) * 32.0f, 0.0f), 31.0f));
    int lin = (int)(fx * 1024.0f + fy * 32.0f + fz);
    float mask = 1.0f - v_b[lin];
    const float* g = cps_b + 3 * lin;
    float dx = (qx - g[0]) * mask;
    float dy = (qy - g[1]) * mask;
    float dz = (qz - g[2]) * mask;
    float n2 = dx * dx + dy * dy + dz * dz;
    return sqrtf(fmaxf(n2, 1e-30f));
}

__global__ __launch_bounds__(256)
void sym_loss_main(const float* __restrict__ planes,
                   const float* __restrict__ quats,
                   const float* __restrict__ cps,
                   const float* __restrict__ samples,
                   const float* __restrict__ vol,
                   float* __restrict__ ws,
                   int B, int N, int S) {
    const int b   = blockIdx.x;
    const int s   = blockIdx.y;
    const int tid = threadIdx.x;

    // uniform per-block params -> scalar loads
    const float* pl = planes + ((size_t)s * B + b) * 4;
    const float nx = pl[0], ny = pl[1], nz = pl[2], pd = pl[3];
    const float den = nx * nx + ny * ny + nz * nz + 1e-8f;

    const float* qv = quats + ((size_t)s * B + b) * 4;
    const float qw = qv[0], qx = qv[1], qy = qv[2], qz = qv[3];

    const float* cps_b = cps     + (size_t)b * (GRID3 * 3);
    const float* v_b   = vol     + (size_t)b * GRID3;
    const float* pts   = samples + (size_t)b * N * 3;

    float acc_p = 0.0f, acc_q = 0.0f;
    for (int i = tid; i < N; i += 256) {
        const float px = pts[3 * i + 0];
        const float py = pts[3 * i + 1];
        const float pz = pts[3 * i + 2];

        // ---- plane reflection: p' = p - 2(p.n + d) * n / (|n|^2 + 1e-8)
        const float len2 = 2.0f * (px * nx + py * ny + pz * nz + pd);
        const float rx = px - (len2 * nx) / den;
        const float ry = py - (len2 * ny) / den;
        const float rz = pz - (len2 * nz) / den;
        acc_p += gather_norm(cps_b, v_b, rx, ry, rz);

        // ---- quaternion rotation: (q * (0,p) * conj(q)).xyz
        const float tw = -(qx * px + qy * py + qz * pz);
        const float tx = qw * px + qy * pz - qz * py;
        const float ty = qw * py - qx * pz + qz * px;
        const float tz = qw * pz + qx * py - qy * px;
        const float ox = -tw * qx + tx * qw - ty * qz + tz * qy;
        const float oy = -tw * qy + tx * qz + ty * qw - tz * qx;
        const float oz = -tw * qz - tx * qy + ty * qx + tz * qw;
        acc_q += gather_norm(cps_b, v_b, ox, oy, oz);
    }

    // ---- block reduction of 256 partials via WMMA (A = ones column-sum trick)
    __shared__ float s_p[256];
    __shared__ float s_q[256];
    s_p[tid] = acc_p;
    s_q[tid] = acc_q;
    __syncthreads();

    if (tid < 32) {  // exactly wave 0: EXEC all ones for the WMMAs
        v2f a;  a.x = 1.0f; a.y = 1.0f;       // A(16x4) = all ones
        v8f cp = {};
        v8f cq = {};
#pragma unroll
        for (int j = 0; j < 4; ++j) {
            v2f bp, bq;                        // B(4x16): 64 partials per WMMA
            bp.x = s_p[j * 64 + tid];
            bp.y = s_p[j * 64 + 32 + tid];
            bq.x = s_q[j * 64 + tid];
            bq.y = s_q[j * 64 + 32 + tid];
            // D[m,n] = sum_k B[k,n] + C[m,n]  (chained accumulation)
            cp = __builtin_amdgcn_wmma_f32_16x16x4_f32(
                    false, a, false, bp, (short)0, cp, false, false);
            cq = __builtin_amdgcn_wmma_f32_16x16x4_f32(
                    false, a, false, bq, (short)0, cq, false, false);
        }
        // Row M=0 of D lives in VGPR0 across lanes 0..15 (N = lane).
        float rp = cp[0];
        float rq = cq[0];
#pragma unroll
        for (int m = 8; m >= 1; m >>= 1) {     // sum the 16 column-sums
            rp += __shfl_xor(rp, m, 32);
            rq += __shfl_xor(rq, m, 32);
        }
        if (tid == 0) {
            ws[(size_t)s * B + b]                   = rp;   // plane partial
            ws[(size_t)S * B + (size_t)s * B + b]   = rq;   // quat partial
        }
    }
}

__global__ __launch_bounds__(256)
void sym_loss_reduce(const float* __restrict__ ws, float* __restrict__ out,
                     int count) {
    __shared__ float sm[256];
    const int tid = threadIdx.x;
    const float* src = ws + (size_t)blockIdx.x * count;
    float acc = 0.0f;
    for (int i = tid; i < count; i += 256) acc += src[i];
    sm[tid] = acc;
    __syncthreads();
#pragma unroll
    for (int off = 128; off >= 1; off >>= 1) {
        if (tid < off) sm[tid] += sm[tid + off];
        __syncthreads();
    }
    if (tid == 0) out[blockIdx.x] = sm[0] / (float)count;  // mean over S*B
}

extern "C" void kernel_launch(void* const* d_in, const int* in_sizes, int n_in,
                              void* d_out, int out_size, void* d_ws, size_t ws_size,
                              hipStream_t stream) {
    const float* planes  = (const float*)d_in[0];  // (S, B, 4)
    const float* quats   = (const float*)d_in[1];  // (S, B, 4)
    const float* cps     = (const float*)d_in[2];  // (B, 32768*3)
    const float* samples = (const float*)d_in[3];  // (B, N, 3)
    const float* vol     = (const float*)d_in[4];  // (B, 1, 32, 32, 32)

    const int B = in_sizes[4] / GRID3;
    const int S = in_sizes[0] / (B * 4);
    const int N = in_sizes[3] / (B * 3);

    float* ws = (float*)d_ws;   // 2*S*B floats of partial sums

    dim3 grid(B, S);
    sym_loss_main<<<grid, 256, 0, stream>>>(planes, quats, cps, samples, vol,
                                            ws, B, N, S);
    // out[0] = mean plane loss, out[1] = mean quat loss
    sym_loss_reduce<<<2, 256, 0, stream>>>(ws, (float*)d_out, S * B);
}